// LimbEmbedding_69896297775214
// MI455X (gfx1250) — compile-verified
//
#include <hip/hip_runtime.h>

typedef __attribute__((ext_vector_type(2))) float v2f;
typedef __attribute__((ext_vector_type(8))) float v8f;

#define D_MODEL 256
#define DH 128
#define NJ 17
#define LIMBS 16
#define WAVES_PER_BLOCK 2
#define TOKENS_PER_WAVE 16
#define LDS_TOKEN_STRIDE 260   // 256 + 4 floats pad: halves hit disjoint LDS banks

__global__ __launch_bounds__(WAVES_PER_BLOCK * 32)
void limb_embed_kernel(const float* __restrict__ xyz,
                       const int*   __restrict__ parent,
                       const int*   __restrict__ child,
                       const float* __restrict__ dir_w,   // [3,128] row-major
                       const float* __restrict__ dir_b,   // [128]
                       const float* __restrict__ len_w,   // [1,128]
                       const float* __restrict__ len_b,   // [128]
                       const float* __restrict__ ln_g,    // [256]
                       const float* __restrict__ ln_beta, // [256]
                       float* __restrict__ out,
                       int n_frames)
{
    __shared__ float lds[WAVES_PER_BLOCK * TOKENS_PER_WAVE * LDS_TOKEN_STRIDE];

    const int lane        = threadIdx.x & 31;
    const int waveInBlock = threadIdx.x >> 5;
    const int frame       = blockIdx.x * WAVES_PER_BLOCK + waveInBlock;
    if (frame >= n_frames) return;   // wave-uniform; terminated waves don't block barriers

    const bool hi = (lane >> 4) != 0; // false: lanes 0-15, true: lanes 16-31
    const int  nl = lane & 15;        // limb index == column-in-tile == token-in-frame

    // ---- gather limb endpoints, build unit vector + length (each half duplicates) ----
    const int pj = parent[nl];
    const int cj = child[nl];
    const float* fb = xyz + (size_t)frame * (NJ * 3);
    const float vx = fb[cj * 3 + 0] - fb[pj * 3 + 0];
    const float vy = fb[cj * 3 + 1] - fb[pj * 3 + 1];
    const float vz = fb[cj * 3 + 2] - fb[pj * 3 + 2];
    float len = sqrtf(vx * vx + vy * vy + vz * vz);
    len = fmaxf(len, 1e-6f);
    const float inv = 1.0f / len;
    const float ux = vx * inv, uy = vy * inv, uz = vz * inv;

    // ---- A operand: 16x4 fp32 layout (K0,K1 in lanes 0-15; K2,K3(=0) in lanes 16-31) ----
    v2f a;
    a.x = hi ? uz : ux;
    a.y = hi ? 0.0f : uy;

    // ---- 8 WMMA tiles covering DH=128 columns; C pre-loaded with dir_b ----
    // Load all three dir_w rows unconditionally so the half-select becomes a
    // register v_cndmask instead of exec-masked predicated loads.
    v8f d[8];
    float lw[8], lb[8], gdir[8], bdir[8], glen[8], blen[8];
#pragma unroll
    for (int t = 0; t < 8; ++t) {
        const int col = t * 16 + nl;
        const float w0 = dir_w[0 * DH + col];
        const float w1 = dir_w[1 * DH + col];
        const float w2 = dir_w[2 * DH + col];
        v2f b;                               // B 4x16: K0,K2 split across halves
        b.x = hi ? w2 : w0;
        b.y = hi ? 0.0f : w1;
        const float bias = dir_b[col];
        v8f c = {bias, bias, bias, bias, bias, bias, bias, bias};
        d[t] = __builtin_amdgcn_wmma_f32_16x16x4_f32(
            /*neg_a=*/false, a, /*neg_b=*/false, b,
            /*c_mod=*/(short)0, c, /*reuse_a=*/false, /*reuse_b=*/false);
        lw[t]   = len_w[col];
        lb[t]   = len_b[col];
        gdir[t] = ln_g[col];
        bdir[t] = ln_beta[col];
        glen[t] = ln_g[DH + col];
        blen[t] = ln_beta[DH + col];
    }

    // ---- broadcast each token's length to all lanes of its half ----
    const int halfOff = hi ? 8 : 0;
    float lenM[8];
#pragma unroll
    for (int r = 0; r < 8; ++r)
        lenM[r] = __shfl(len, r + halfOff, 32);

    // ---- LayerNorm statistics per token (vgpr slot r <-> token r + 8*half) ----
    float mu[8], rs[8];
#pragma unroll
    for (int r = 0; r < 8; ++r) {
        float s = 0.0f, s2 = 0.0f;
#pragma unroll
        for (int t = 0; t < 8; ++t) {
            const float xd = d[t][r];
            const float xl = fmaf(lenM[r], lw[t], lb[t]);
            s  += xd + xl;
            s2 += xd * xd + xl * xl;
        }
#pragma unroll
        for (int m = 1; m < 16; m <<= 1) {   // stays within the 16-lane half
            s  += __shfl_xor(s,  m, 32);
            s2 += __shfl_xor(s2, m, 32);
        }
        const float mean = s * (1.0f / 256.0f);
        const float var  = s2 * (1.0f / 256.0f) - mean * mean;
        mu[r] = mean;
        rs[r] = rsqrtf(var + 1e-5f);
    }

    // ---- normalize + affine, stage to LDS in [token][dim] layout ----
    float* wlds = lds + waveInBlock * TOKENS_PER_WAVE * LDS_TOKEN_STRIDE;
#pragma unroll
    for (int r = 0; r < 8; ++r) {
        const int tok = r + halfOff;
        float* row = wlds + tok * LDS_TOKEN_STRIDE;
#pragma unroll
        for (int t = 0; t < 8; ++t) {
            const float xd = (d[t][r] - mu[r]) * rs[r];
            row[t * 16 + nl] = fmaf(xd, gdir[t], bdir[t]);
            const float xl = (fmaf(lenM[r], lw[t], lb[t]) - mu[r]) * rs[r];
            row[DH + t * 16 + nl] = fmaf(xl, glen[t], blen[t]);
        }
    }
    __syncthreads();

    // ---- coalesced write-out: 512B contiguous per global_store_b128 ----
    const size_t outBase = (size_t)frame * (TOKENS_PER_WAVE * D_MODEL);
#pragma unroll
    for (int i = 0; i < 32; ++i) {
        const int tok = i >> 1;
        const int dim = ((i & 1) << 7) + lane * 4;
        const float4 val = *(const float4*)(wlds + tok * LDS_TOKEN_STRIDE + dim);
        *(float4*)(out + outBase + (size_t)tok * D_MODEL + dim) = val;
    }
}

extern "C" void kernel_launch(void* const* d_in, const int* in_sizes, int n_in,
                              void* d_out, int out_size, void* d_ws, size_t ws_size,
                              hipStream_t stream) {
    const float* xyz     = (const float*)d_in[0];
    const int*   parent  = (const int*)d_in[1];
    const int*   child   = (const int*)d_in[2];
    const float* dir_w   = (const float*)d_in[3];
    const float* dir_b   = (const float*)d_in[4];
    const float* len_w   = (const float*)d_in[5];
    const float* len_b   = (const float*)d_in[6];
    const float* ln_g    = (const float*)d_in[7];
    const float* ln_beta = (const float*)d_in[8];

    const int frames = in_sizes[0] / (NJ * 3);   // B*T = 16384
    const int blocks = (frames + WAVES_PER_BLOCK - 1) / WAVES_PER_BLOCK;

    limb_embed_kernel<<<blocks, WAVES_PER_BLOCK * 32, 0, stream>>>(
        xyz, parent, child, dir_w, dir_b, len_w, len_b, ln_g, ln_beta,
        (float*)d_out, frames);
}